// MaceNet_5153960755637
// MI455X (gfx1250) — compile-verified
//
#include <hip/hip_runtime.h>
#include <math.h>

#define N_NODES 512
#define F_DIM   128
#define V_DIM   64
#define B_DIM   10
#define FR_DIM  64
#define VR_DIM  8
#define R_MAX_F 10.0f
#define PI_F    3.14159265358979323846f

typedef __attribute__((ext_vector_type(2))) float v2f;
typedef __attribute__((ext_vector_type(8))) float v8f;

// ---------------------------------------------------------------------------
// Edge aggregation: one workgroup per receiver node i.
// Threads 0..127   -> scalar-channel path (feature f = tid)
// Threads 128..191 -> vector-channel path (channel v = tid-128)
// The Wmix_B / Wmix_C linear maps are pulled OUT of the edge loop (linearity)
// and applied later as WMMA GEMMs -> ~6x fewer edge FLOPs than the naive form.
// ---------------------------------------------------------------------------
__global__ __launch_bounds__(192)
void edge_kernel(const float* __restrict__ x,
                 const float* __restrict__ s_in,   // [N,F]
                 const float* __restrict__ v_in,   // [N,3,V]
                 const float* __restrict__ WrA, const float* __restrict__ WrB,
                 const float* __restrict__ WrC, const float* __restrict__ WrD,
                 const float* __restrict__ WrE,
                 float* __restrict__ accA_out,     // [N,F]   sum_j s_j*wA
                 float* __restrict__ tB_out,       // [N,V]   sum_j (v_j.y1)*wB
                 float* __restrict__ P_out,        // [N,3,F] sum_j s_j*wC*y1_d
                 float* __restrict__ DE_out)       // [N,3,V] sum_j v_j*wD + (v_j x y1)*wE
{
    __shared__ float wra[B_DIM*F_DIM];
    __shared__ float wrc[B_DIM*F_DIM];
    __shared__ float wrb[B_DIM*V_DIM];
    __shared__ float wrd[B_DIM*V_DIM];
    __shared__ float wre[B_DIM*V_DIM];
    __shared__ float rb_s[192*B_DIM];
    __shared__ float y1_s[192*3];

    const int i = blockIdx.x;
    const int t = threadIdx.x;

    for (int k = t; k < B_DIM*F_DIM; k += 192) { wra[k] = WrA[k]; wrc[k] = WrC[k]; }
    for (int k = t; k < B_DIM*V_DIM; k += 192) { wrb[k] = WrB[k]; wrd[k] = WrD[k]; wre[k] = WrE[k]; }
    __syncthreads();

    const float xi0 = x[i*3+0], xi1 = x[i*3+1], xi2 = x[i*3+2];
    const float bes_c = sqrtf(2.0f / R_MAX_F);

    float accA = 0.f, accP0 = 0.f, accP1 = 0.f, accP2 = 0.f;   // scalar path
    float accT = 0.f, accD0 = 0.f, accD1 = 0.f, accD2 = 0.f;   // vector path

    for (int base = 0; base < N_NODES - 1; base += 192) {
        int rem = (N_NODES - 1) - base;
        int cnt = rem < 192 ? rem : 192;

        // Stage per-edge scalars (r, y1, enveloped Bessel features) for this chunk.
        int k = base + t;
        if (k < N_NODES - 1) {
            int j = k + (k >= i ? 1 : 0);                       // senders = all nodes != i
            // warm L2/WGP$ for the rows the whole block will stream next
            __builtin_prefetch(&s_in[j*F_DIM], 0, 1);           // -> global_prefetch_b8
            __builtin_prefetch(&v_in[j*3*V_DIM], 0, 1);
            float d0 = xi0 - x[j*3+0];
            float d1 = xi1 - x[j*3+1];
            float d2 = xi2 - x[j*3+2];
            float r = sqrtf(d0*d0 + d1*d1 + d2*d2);
            float inv_r = 1.0f / r;
            y1_s[t*3+0] = d0*inv_r; y1_s[t*3+1] = d1*inv_r; y1_s[t*3+2] = d2*inv_r;
            float env = (r < R_MAX_F) ? 0.5f*(cosf(PI_F*r/R_MAX_F) + 1.0f) : 0.0f;
            float cs = bes_c * inv_r * env;
            #pragma unroll
            for (int b = 0; b < B_DIM; b++)
                rb_s[t*B_DIM+b] = cs * sinf((float)(b+1) * (PI_F/R_MAX_F) * r);
        }
        __syncthreads();

        for (int jj = 0; jj < cnt; jj++) {
            int kk = base + jj;
            int j = kk + (kk >= i ? 1 : 0);
            float y10 = y1_s[jj*3+0], y11 = y1_s[jj*3+1], y12 = y1_s[jj*3+2];
            const float* rb = &rb_s[jj*B_DIM];
            if (t < F_DIM) {                                   // waves 0..3: scalar channels
                float sv = s_in[j*F_DIM + t];
                float wA = 0.f, wC = 0.f;
                #pragma unroll
                for (int b = 0; b < B_DIM; b++) {
                    float rv = rb[b];
                    wA = fmaf(rv, wra[b*F_DIM+t], wA);
                    wC = fmaf(rv, wrc[b*F_DIM+t], wC);
                }
                accA = fmaf(sv, wA, accA);
                float sc = sv * wC;
                accP0 = fmaf(sc, y10, accP0);
                accP1 = fmaf(sc, y11, accP1);
                accP2 = fmaf(sc, y12, accP2);
            } else {                                           // waves 4..5: vector channels
                int vi = t - F_DIM;
                float va = v_in[(j*3+0)*V_DIM + vi];
                float vb = v_in[(j*3+1)*V_DIM + vi];
                float vc = v_in[(j*3+2)*V_DIM + vi];
                float wB = 0.f, wD = 0.f, wE = 0.f;
                #pragma unroll
                for (int b = 0; b < B_DIM; b++) {
                    float rv = rb[b];
                    wB = fmaf(rv, wrb[b*V_DIM+vi], wB);
                    wD = fmaf(rv, wrd[b*V_DIM+vi], wD);
                    wE = fmaf(rv, wre[b*V_DIM+vi], wE);
                }
                float dotv = va*y10 + vb*y11 + vc*y12;
                accT = fmaf(dotv, wB, accT);
                float c0 = vb*y12 - vc*y11;
                float c1 = vc*y10 - va*y12;
                float c2 = va*y11 - vb*y10;
                accD0 += va*wD + c0*wE;
                accD1 += vb*wD + c1*wE;
                accD2 += vc*wD + c2*wE;
            }
        }
        __syncthreads();
    }

    if (t < F_DIM) {
        accA_out[i*F_DIM + t]      = accA;
        P_out[(i*3+0)*F_DIM + t]   = accP0;
        P_out[(i*3+1)*F_DIM + t]   = accP1;
        P_out[(i*3+2)*F_DIM + t]   = accP2;
    } else {
        int vi = t - F_DIM;
        tB_out[i*V_DIM + vi]       = accT;
        DE_out[(i*3+0)*V_DIM + vi] = accD0;
        DE_out[(i*3+1)*V_DIM + vi] = accD1;
        DE_out[(i*3+2)*V_DIM + vi] = accD2;
    }
}

// ---------------------------------------------------------------------------
// fp32 WMMA GEMM v2: OUT = epilogue( A@W [+ A2@W2] [+ addC] ).
// Block = 4 waves = 64x64 output tile; wave w owns the 16x64 strip starting at
// row 16*w, holding 4 v8f accumulators. The 64-column weight panel is staged
// in LDS pair-interleaved (lds[(k/2)*64+n] holds {W[k][n], W[k+1][n]}) so the
// B fragment is a single aligned ds_load_b64; the A fragment (global b64) is
// reused across 4 WMMAs.  Requires M%64==0, N%64==0, K%4==0, K<=128.
// VGPR layouts per CDNA5 ISA 7.12.2:
//   A 16x4 f32 : lane L -> row m=L&15, K pair (2*(L>>4), 2*(L>>4)+1)
//   B  4x16 f32: lane L -> col n=L&15, K pair (2*(L>>4), 2*(L>>4)+1)
//   C/D        : lane L -> col n=L&15, vgpr r -> row r + 8*(L>>4)
// mode: 0 = scale only, 1 = silu, 2 = sigmoid, 3 = * gate[(row/3)*N + col]
// ---------------------------------------------------------------------------
__global__ __launch_bounds__(128)
void gemm_wmma(const float* __restrict__ A,  const float* __restrict__ W,
               const float* __restrict__ A2, const float* __restrict__ W2,
               const float* __restrict__ addC, const float* __restrict__ gate,
               float* __restrict__ out,
               int M, int K, int N, float scale, int mode)
{
    __shared__ float ldsb[128 * 64];          // pair-swizzled K x 64 panel (32 KB)

    const int t     = threadIdx.x;
    const int lane  = t & 31;
    const int wave  = t >> 5;
    const int nblks = N >> 6;
    const int mblk  = (int)blockIdx.x / nblks;
    const int nblk  = (int)blockIdx.x % nblks;
    const int ncol0 = nblk << 6;

    const int half = lane >> 4;               // 0 or 1
    const int mrow = lane & 15;
    const int ncol = lane & 15;
    const int m0   = (mblk << 6) + (wave << 4);

    const float* arow = A + (m0 + mrow) * K;
    const float* a2row = A2 ? A2 + (m0 + mrow) * K : nullptr;

    // ---- stage W panel (pair-interleaved) ----
    for (int idx = t; idx < K * 64; idx += 128) {
        int k = idx >> 6, n = idx & 63;
        ldsb[(((k >> 1) << 6) + n) * 2 + (k & 1)] = W[k * N + ncol0 + n];
    }
    if (a2row) __builtin_prefetch(a2row, 0, 1);   // warm A2 row for pass 2
    __syncthreads();

    v8f acc0 = {0,0,0,0,0,0,0,0}, acc1 = acc0, acc2 = acc0, acc3 = acc0;

    for (int k0 = 0; k0 < K; k0 += 4) {
        v2f a = *(const v2f*)(arow + k0 + 2*half);
        const float* bp = &ldsb[(((k0 >> 1) + half) << 6) * 2];
        v2f b0 = *(const v2f*)(bp + (ncol      ) * 2);
        v2f b1 = *(const v2f*)(bp + (ncol + 16 ) * 2);
        v2f b2 = *(const v2f*)(bp + (ncol + 32 ) * 2);
        v2f b3 = *(const v2f*)(bp + (ncol + 48 ) * 2);
        acc0 = __builtin_amdgcn_wmma_f32_16x16x4_f32(false, a, false, b0, (short)0, acc0, false, false);
        acc1 = __builtin_amdgcn_wmma_f32_16x16x4_f32(false, a, false, b1, (short)0, acc1, false, false);
        acc2 = __builtin_amdgcn_wmma_f32_16x16x4_f32(false, a, false, b2, (short)0, acc2, false, false);
        acc3 = __builtin_amdgcn_wmma_f32_16x16x4_f32(false, a, false, b3, (short)0, acc3, false, false);
    }

    if (a2row) {                               // fused second GEMM: re-stage panel
        __syncthreads();
        for (int idx = t; idx < K * 64; idx += 128) {
            int k = idx >> 6, n = idx & 63;
            ldsb[(((k >> 1) << 6) + n) * 2 + (k & 1)] = W2[k * N + ncol0 + n];
        }
        __syncthreads();
        for (int k0 = 0; k0 < K; k0 += 4) {
            v2f a = *(const v2f*)(a2row + k0 + 2*half);
            const float* bp = &ldsb[(((k0 >> 1) + half) << 6) * 2];
            v2f b0 = *(const v2f*)(bp + (ncol      ) * 2);
            v2f b1 = *(const v2f*)(bp + (ncol + 16 ) * 2);
            v2f b2 = *(const v2f*)(bp + (ncol + 32 ) * 2);
            v2f b3 = *(const v2f*)(bp + (ncol + 48 ) * 2);
            acc0 = __builtin_amdgcn_wmma_f32_16x16x4_f32(false, a, false, b0, (short)0, acc0, false, false);
            acc1 = __builtin_amdgcn_wmma_f32_16x16x4_f32(false, a, false, b1, (short)0, acc1, false, false);
            acc2 = __builtin_amdgcn_wmma_f32_16x16x4_f32(false, a, false, b2, (short)0, acc2, false, false);
            acc3 = __builtin_amdgcn_wmma_f32_16x16x4_f32(false, a, false, b3, (short)0, acc3, false, false);
        }
    }

    // ---- epilogue ----
    v8f accs[4] = {acc0, acc1, acc2, acc3};
    #pragma unroll
    for (int nt = 0; nt < 4; nt++) {
        #pragma unroll
        for (int r = 0; r < 8; r++) {
            int row = m0 + r + 8*half;
            int col = ncol0 + nt*16 + ncol;
            float y = accs[nt][r];
            if (addC) y += addC[row * N + col];
            y *= scale;
            if (mode == 1)      y = y / (1.0f + expf(-y));       // silu
            else if (mode == 2) y = 1.0f / (1.0f + expf(-y));    // sigmoid
            else if (mode == 3) y *= gate[(row / 3) * N + col];  // gated v-update
            out[row * N + col] = y;
        }
    }
}

// ---------------------------------------------------------------------------
__global__ void init_kernel(const float* __restrict__ embed,
                            float* __restrict__ s0, float* __restrict__ v0)
{
    int idx = blockIdx.x * blockDim.x + threadIdx.x;
    if (idx < N_NODES * F_DIM)     s0[idx] = embed[idx & (F_DIM - 1)];
    if (idx < N_NODES * 3 * V_DIM) v0[idx] = 0.0f;
}

__global__ void com_kernel(const float* __restrict__ x, float* __restrict__ com)
{
    int d = threadIdx.x;
    if (d < 3) {
        float s = 0.f;
        for (int n = 0; n < N_NODES; n++) s += x[n*3 + d];
        com[d] = s / (float)N_NODES;
    }
}

__global__ void vec_out_kernel(const float* __restrict__ v_fin,   // [N,3,V]
                               const float* __restrict__ Wv,      // [V,VR]
                               const float* __restrict__ com,
                               float* __restrict__ out)           // [N,VR,3]
{
    int idx = blockIdx.x * blockDim.x + threadIdx.x;
    if (idx >= N_NODES * VR_DIM * 3) return;
    int n   = idx / (VR_DIM * 3);
    int rem = idx % (VR_DIM * 3);
    int rr  = rem / 3;
    int d   = rem % 3;
    const float* vp = v_fin + (n*3 + d) * V_DIM;
    float acc = 0.f;
    #pragma unroll 8
    for (int vv = 0; vv < V_DIM; vv++) acc = fmaf(vp[vv], Wv[vv*VR_DIM + rr], acc);
    out[idx] = acc + com[d];
}

// ---------------------------------------------------------------------------
extern "C" void kernel_launch(void* const* d_in, const int* in_sizes, int n_in,
                              void* d_out, int out_size, void* d_ws, size_t ws_size,
                              hipStream_t stream)
{
    (void)in_sizes; (void)n_in; (void)out_size; (void)ws_size;
    // setup_inputs() insertion order:
    // 0:x 1:embed | per layer l (base 2+12l): Wr_A Wr_B Wr_C Wr_D Wr_E
    //   Wmix_B Wmix_C Ws_agg Ws_self Wv_agg Wv_self W_gate | 26:W_out_s 27:W_out_v
    const float* x     = (const float*)d_in[0];
    const float* embed = (const float*)d_in[1];
    const float* Wouts = (const float*)d_in[26];
    const float* Woutv = (const float*)d_in[27];

    float* ws   = (float*)d_ws;
    float* s0   = ws; ws += N_NODES * F_DIM;
    float* s1   = ws; ws += N_NODES * F_DIM;
    float* v0   = ws; ws += N_NODES * 3 * V_DIM;
    float* v1   = ws; ws += N_NODES * 3 * V_DIM;
    float* accA = ws; ws += N_NODES * F_DIM;
    float* tB   = ws; ws += N_NODES * V_DIM;
    float* P    = ws; ws += N_NODES * 3 * F_DIM;
    float* DE   = ws; ws += N_NODES * 3 * V_DIM;
    float* aggS = ws; ws += N_NODES * F_DIM;
    float* aggV = ws; ws += N_NODES * 3 * V_DIM;
    float* gate = ws; ws += N_NODES * V_DIM;
    float* com  = ws; ws += 4;

    const float inv_avg = 1.0f / 511.0f;

    init_kernel<<<(N_NODES*3*V_DIM + 255)/256, 256, 0, stream>>>(embed, s0, v0);
    com_kernel<<<1, 32, 0, stream>>>(x, com);

    float* sbuf[2] = {s0, s1};
    float* vbuf[2] = {v0, v1};

    for (int l = 0; l < 2; l++) {
        const int b = 2 + l * 12;
        const float* WrA    = (const float*)d_in[b + 0];
        const float* WrB    = (const float*)d_in[b + 1];
        const float* WrC    = (const float*)d_in[b + 2];
        const float* WrD    = (const float*)d_in[b + 3];
        const float* WrE    = (const float*)d_in[b + 4];
        const float* WmixB  = (const float*)d_in[b + 5];
        const float* WmixC  = (const float*)d_in[b + 6];
        const float* Wsagg  = (const float*)d_in[b + 7];
        const float* Wsself = (const float*)d_in[b + 8];
        const float* Wvagg  = (const float*)d_in[b + 9];
        const float* Wvself = (const float*)d_in[b + 10];
        const float* Wgate  = (const float*)d_in[b + 11];

        float* s_in  = sbuf[l & 1];
        float* s_out = sbuf[(l + 1) & 1];
        float* v_in  = vbuf[l & 1];
        float* v_out = vbuf[(l + 1) & 1];

        edge_kernel<<<N_NODES, 192, 0, stream>>>(x, s_in, v_in,
                                                 WrA, WrB, WrC, WrD, WrE,
                                                 accA, tB, P, DE);
        // aggS = (accA + tB @ Wmix_B) / 511        [512x64]@[64x128]
        gemm_wmma<<<(512/64)*(128/64), 128, 0, stream>>>(
            tB, WmixB, nullptr, nullptr, accA, nullptr, aggS, 512, 64, 128, inv_avg, 0);
        // s_out = silu(aggS @ Ws_agg + s_in @ Ws_self)   2x [512x128]@[128x128]
        gemm_wmma<<<(512/64)*(128/64), 128, 0, stream>>>(
            aggS, Wsagg, s_in, Wsself, nullptr, nullptr, s_out, 512, 128, 128, 1.0f, 1);
        // gate = sigmoid(s_out @ W_gate)           [512x128]@[128x64]
        gemm_wmma<<<(512/64)*(64/64), 128, 0, stream>>>(
            s_out, Wgate, nullptr, nullptr, nullptr, nullptr, gate, 512, 128, 64, 1.0f, 2);
        // aggV = (P @ Wmix_C + DE) / 511           [1536x128]@[128x64]
        gemm_wmma<<<(1536/64)*(64/64), 128, 0, stream>>>(
            P, WmixC, nullptr, nullptr, DE, nullptr, aggV, 1536, 128, 64, inv_avg, 0);
        // v_out = (aggV @ Wv_agg + v_in @ Wv_self) * gate   2x [1536x64]@[64x64]
        gemm_wmma<<<(1536/64)*(64/64), 128, 0, stream>>>(
            aggV, Wvagg, v_in, Wvself, nullptr, gate, v_out, 1536, 64, 64, 1.0f, 3);
    }

    float* s_fin = sbuf[0];   // 2 layers: s0 -> s1 -> s0
    float* v_fin = vbuf[0];

    // inv = s_fin @ W_out_s  -> second output region   [512x128]@[128x64]
    gemm_wmma<<<(512/64)*(64/64), 128, 0, stream>>>(
        s_fin, Wouts, nullptr, nullptr, nullptr, nullptr,
        (float*)d_out + N_NODES*VR_DIM*3, 512, 128, 64, 1.0f, 0);
    // vecs = v_fin @ W_out_v + com -> first output region
    vec_out_kernel<<<(N_NODES*VR_DIM*3 + 255)/256, 256, 0, stream>>>(
        v_fin, Woutv, com, (float*)d_out);
}